// GCNProbe_52682068853004
// MI455X (gfx1250) — compile-verified
//
#include <hip/hip_runtime.h>

// ---------------------------------------------------------------------------
// GCN forward for MI455X (gfx1250, wave32).
// Memory-bound workload: node tensors (25.6 MB) are L2-resident (192 MB L2),
// edge gather/scatter resolves in L2 via global_atomic_add_f32.
// Dense 64x64 transforms use V_WMMA_F32_16X16X4_F32 (exact f32 path).
// ---------------------------------------------------------------------------

#define N_NODES  100000
#define N_EDGES  1600000
#define HIDDEN   64
#define N_LAYERS 3
#define N_GRAPHS 64

typedef __attribute__((ext_vector_type(2))) float v2f;
typedef __attribute__((ext_vector_type(8))) float v8f;

#if defined(__has_builtin)
#if __has_builtin(__builtin_amdgcn_wmma_f32_16x16x4_f32)
#define HAVE_WMMA_F32 1
#endif
#endif

// ---------------------------------------------------------------------------
// h[i][j] = emb[x[i]][j]
__global__ __launch_bounds__(256) void gcn_gather_emb(
    const int* __restrict__ x, const float* __restrict__ emb,
    float* __restrict__ h) {
  int idx = blockIdx.x * 256 + threadIdx.x;      // grid covers N_NODES*HIDDEN
  int node = idx >> 6;
  int feat = idx & 63;
  if (node < N_NODES) {
    h[(size_t)node * HIDDEN + feat] = emb[(size_t)x[node] * HIDDEN + feat];
  }
}

// ---------------------------------------------------------------------------
// ht = (relu?)(hin) @ W   for a 64x64 W.  One wave computes a 16x64 strip.
// K=64 swept in 16 steps of 4 via V_WMMA_F32_16X16X4_F32, 4 N-tiles of 16.
//
// A 16x4 f32 layout (ISA 7.12.2): lanes 0-15 M=lane, VGPR0=K(kb), VGPR1=K(kb+1)
// with kb = (lane>>4)*2.  B 4x16 mirrored: N=lane&15.  C/D: VGPR r holds
// M = r + (lane>>4)*8, N = lane&15.
__global__ __launch_bounds__(256) void gcn_transform_wmma(
    const float* __restrict__ hin, const float* __restrict__ W,
    float* __restrict__ ht, int relu_in) {
  __shared__ float Wl[HIDDEN * HIDDEN];          // 16 KB weight stage
  const int tid = threadIdx.x;
  for (int i = tid; i < HIDDEN * HIDDEN; i += 256) Wl[i] = W[i];
  __syncthreads();

  const int wave = tid >> 5;
  const int lane = tid & 31;
  const int tile = blockIdx.x * 8 + wave;        // 16-row tile index
  const int n_tiles = N_NODES / 16;              // 6250 exactly
  if (tile >= n_tiles) return;

  const int row0 = tile * 16;
  const int m    = lane & 15;
  const int kb   = (lane >> 4) * 2;

  v8f acc[4];
#pragma unroll
  for (int nt = 0; nt < 4; ++nt) acc[nt] = (v8f){0,0,0,0,0,0,0,0};

#if HAVE_WMMA_F32
  const float* __restrict__ arow = hin + (size_t)(row0 + m) * HIDDEN;
  for (int k = 0; k < HIDDEN; k += 4) {
    float a0 = arow[k + kb];
    float a1 = arow[k + kb + 1];
    if (relu_in) { a0 = fmaxf(a0, 0.f); a1 = fmaxf(a1, 0.f); }
    v2f a; a.x = a0; a.y = a1;
#pragma unroll
    for (int nt = 0; nt < 4; ++nt) {
      const int n = (lane & 15) + nt * 16;
      v2f b;
      b.x = Wl[(k + kb)     * HIDDEN + n];
      b.y = Wl[(k + kb + 1) * HIDDEN + n];
      acc[nt] = __builtin_amdgcn_wmma_f32_16x16x4_f32(
          /*neg_a=*/false, a, /*neg_b=*/false, b,
          /*c_mod=*/(short)0, acc[nt], /*reuse_a=*/false, /*reuse_b=*/false);
    }
  }
#else
  // Scalar fallback (compile-safe); histogram will show wmma==0 if taken.
  for (int nt = 0; nt < 4; ++nt) {
#pragma unroll
    for (int r = 0; r < 8; ++r) {
      const int mm = r + (lane >> 4) * 8;
      const int n  = (lane & 15) + nt * 16;
      float s = 0.f;
      for (int k = 0; k < HIDDEN; ++k) {
        float av = hin[(size_t)(row0 + mm) * HIDDEN + k];
        if (relu_in) av = fmaxf(av, 0.f);
        s += av * Wl[k * HIDDEN + n];
      }
      acc[nt][r] = s;
    }
  }
#endif

#pragma unroll
  for (int nt = 0; nt < 4; ++nt) {
#pragma unroll
    for (int r = 0; r < 8; ++r) {
      const int mm = r + (lane >> 4) * 8;
      ht[(size_t)(row0 + mm) * HIDDEN + nt * 16 + (lane & 15)] = acc[nt][r];
    }
  }
}

// ---------------------------------------------------------------------------
// h_next[i][j] = bias[j]   (bias folded into scatter destination init)
__global__ __launch_bounds__(256) void gcn_init_bias(
    float* __restrict__ hn, const float* __restrict__ b) {
  int idx = blockIdx.x * 256 + threadIdx.x;      // grid covers N_NODES*HIDDEN
  if ((idx >> 6) < N_NODES) hn[idx] = b[idx & 63];
}

// ---------------------------------------------------------------------------
// h_next[dst[e]] += w[e] * ht[src[e]]  — 32 lanes per edge, float2 per lane.
// Atomic RMWs land in the 192 MB L2 (h_next is 25.6 MB, fully resident).
__global__ __launch_bounds__(256) void gcn_edge_scatter(
    const int* __restrict__ src, const int* __restrict__ dst,
    const float* __restrict__ ew, const float* __restrict__ ht,
    float* __restrict__ hn) {
  const int lane = threadIdx.x & 31;
  const int e = blockIdx.x * 8 + (threadIdx.x >> 5);
  if (e >= N_EDGES) return;
  const int s = src[e];
  const int d = dst[e];
  const float w = ew[e];
  const float2 v = ((const float2*)(ht + (size_t)s * HIDDEN))[lane];
  float* drow = hn + (size_t)d * HIDDEN + lane * 2;
  atomicAdd(drow + 0, w * v.x);
  atomicAdd(drow + 1, w * v.y);
}

// ---------------------------------------------------------------------------
__global__ __launch_bounds__(256) void gcn_pool_init(
    float* __restrict__ mean, unsigned* __restrict__ mx,
    float* __restrict__ cnt) {
  int i = blockIdx.x * 256 + threadIdx.x;
  if (i < N_GRAPHS * HIDDEN) { mean[i] = 0.f; mx[i] = 0u; }
  if (i < N_GRAPHS) cnt[i] = 0.f;
}

// Segment mean-sum / max per graph; ReLU of the final layer fused here.
// Post-ReLU values >= 0 so float max == u32 bit-pattern max.
__global__ __launch_bounds__(256) void gcn_pool(
    const float* __restrict__ h, const int* __restrict__ batch,
    float* __restrict__ mean, unsigned* __restrict__ mx,
    float* __restrict__ cnt) {
  const int lane = threadIdx.x & 31;
  const int i = blockIdx.x * 8 + (threadIdx.x >> 5);
  if (i >= N_NODES) return;
  const int g = batch[i];
  float2 v = ((const float2*)(h + (size_t)i * HIDDEN))[lane];
  v.x = fmaxf(v.x, 0.f);
  v.y = fmaxf(v.y, 0.f);
  float* mrow = mean + (size_t)g * HIDDEN + lane * 2;
  atomicAdd(mrow + 0, v.x);
  atomicAdd(mrow + 1, v.y);
  unsigned* xrow = mx + (size_t)g * HIDDEN + lane * 2;
  atomicMax(xrow + 0, __float_as_uint(v.x));
  atomicMax(xrow + 1, __float_as_uint(v.y));
  if (lane == 0) atomicAdd(&cnt[g], 1.0f);
}

// ---------------------------------------------------------------------------
// g = [mean, max]; y = relu(g @ fc1_w + fc1_b); out = y @ fc2_w + fc2_b
__global__ __launch_bounds__(256) void gcn_mlp(
    const float* __restrict__ mean, const unsigned* __restrict__ mx,
    const float* __restrict__ cnt,
    const float* __restrict__ fc1w, const float* __restrict__ fc1b,
    const float* __restrict__ fc2w, const float* __restrict__ fc2b,
    float* __restrict__ out) {
  __shared__ float gf[N_GRAPHS][2 * HIDDEN];     // 32 KB
  __shared__ float y[N_GRAPHS][HIDDEN];          // 16 KB
  const int tid = threadIdx.x;

  for (int idx = tid; idx < N_GRAPHS * 2 * HIDDEN; idx += 256) {
    const int g = idx >> 7;
    const int k = idx & 127;
    float v;
    if (k < HIDDEN) v = mean[g * HIDDEN + k] / fmaxf(cnt[g], 1.0f);
    else            v = __uint_as_float(mx[g * HIDDEN + (k - HIDDEN)]);
    gf[g][k] = v;
  }
  __syncthreads();

  for (int idx = tid; idx < N_GRAPHS * HIDDEN; idx += 256) {
    const int g = idx >> 6;
    const int j = idx & 63;
    float s = fc1b[j];
    for (int k = 0; k < 2 * HIDDEN; ++k) s += gf[g][k] * fc1w[k * HIDDEN + j];
    y[g][j] = fmaxf(s, 0.f);
  }
  __syncthreads();

  if (tid < N_GRAPHS) {
    float s = fc2b[0];
    for (int j = 0; j < HIDDEN; ++j) s += y[tid][j] * fc2w[j];
    out[tid] = s;
  }
}

// ---------------------------------------------------------------------------
extern "C" void kernel_launch(void* const* d_in, const int* in_sizes, int n_in,
                              void* d_out, int out_size, void* d_ws, size_t ws_size,
                              hipStream_t stream) {
  (void)in_sizes; (void)n_in; (void)out_size; (void)ws_size;

  const int*   x      = (const int*)  d_in[0];
  const int*   ei     = (const int*)  d_in[1];  // [2, E]: src then dst
  const float* ew     = (const float*)d_in[2];
  const int*   batch  = (const int*)  d_in[3];
  const float* emb    = (const float*)d_in[4];
  const float* conv_w = (const float*)d_in[5];  // [3, 64, 64]
  const float* conv_b = (const float*)d_in[6];  // [3, 64]
  const float* fc1w   = (const float*)d_in[7];  // [128, 64]
  const float* fc1b   = (const float*)d_in[8];
  const float* fc2w   = (const float*)d_in[9];  // [64, 1]
  const float* fc2b   = (const float*)d_in[10];
  float* out = (float*)d_out;

  // Workspace carve-up (~77 MB)
  const size_t NH = (size_t)N_NODES * HIDDEN;
  float*    hA   = (float*)d_ws;
  float*    hB   = hA + NH;
  float*    hT   = hB + NH;
  float*    mean = hT + NH;
  unsigned* mx   = (unsigned*)(mean + (size_t)N_GRAPHS * HIDDEN);
  float*    cnt  = (float*)(mx + (size_t)N_GRAPHS * HIDDEN);

  const int*   src = ei;
  const int*   dst = ei + N_EDGES;

  // 1) embedding gather
  gcn_gather_emb<<<(N_NODES * HIDDEN) / 256, 256, 0, stream>>>(x, emb, hA);

  // 2) three GCN layers (relu fused into next consumer's load)
  float* cur = hA;
  float* nxt = hB;
  const int tf_blocks   = (N_NODES / 16 + 7) / 8;      // 6250 wave-tiles
  const int init_blocks = (N_NODES * HIDDEN) / 256;
  const int edge_blocks = N_EDGES / 8;
  for (int l = 0; l < N_LAYERS; ++l) {
    gcn_transform_wmma<<<tf_blocks, 256, 0, stream>>>(
        cur, conv_w + (size_t)l * HIDDEN * HIDDEN, hT, l > 0 ? 1 : 0);
    gcn_init_bias<<<init_blocks, 256, 0, stream>>>(nxt, conv_b + (size_t)l * HIDDEN);
    gcn_edge_scatter<<<edge_blocks, 256, 0, stream>>>(src, dst, ew, hT, nxt);
    float* t = cur; cur = nxt; nxt = t;
  }

  // 3) per-graph pooling (relu fused)
  gcn_pool_init<<<(N_GRAPHS * HIDDEN + 255) / 256, 256, 0, stream>>>(mean, mx, cnt);
  gcn_pool<<<N_NODES / 8, 256, 0, stream>>>(cur, batch, mean, mx, cnt);

  // 4) readout MLP
  gcn_mlp<<<1, 256, 0, stream>>>(mean, mx, cnt, fc1w, fc1b, fc2w, fc2b, out);
}